// ATSS_35794257445057
// MI455X (gfx1250) — compile-verified
//
#include <hip/hip_runtime.h>
#include <hip/hip_bf16.h>

#define B_   16
#define N_   30000
#define G_   64
#define K_   9
#define TILES_ (N_/16)     // 1875, exact
#define CACHE_T 128        // tiles cached in LDS per wave (128*32*8B = 32 KB)

typedef __attribute__((ext_vector_type(2))) float v2f;
typedef __attribute__((ext_vector_type(8))) float v8f;

#define KEY_MAX 0x7fffffffffffffffLL

// dist^2 >= 0  =>  float bits order monotonically as integers.
// key = (bits(dist2) << 32) | pred_idx : signed-i64 order == (dist asc, idx asc).
__device__ __forceinline__ long long mkkey(float v, int n) {
    return ((long long)(unsigned long long)__float_as_uint(v) << 32) |
           (unsigned long long)(unsigned int)n;
}

// One 16x16 dist^2 tile via V_WMMA_F32_16X16X4_F32, branchless top-1 update.
__device__ __forceinline__ void tile_update(v2f a, v2f bb, int n,
                                            const long long* __restrict__ prev,
                                            long long* __restrict__ best) {
    v8f c = {};
    c = __builtin_amdgcn_wmma_f32_16x16x4_f32(false, a, false, bb,
                                              (short)0, c, false, false);
#pragma unroll
    for (int v = 0; v < 8; ++v) {
        const long long key = mkkey(c[v], n);
        const bool take = (key > prev[v]) & (key < best[v]);
        best[v] = take ? key : best[v];
    }
}

// One wave (32 threads) per block: 16 GT rows x one chunk of pred tiles.
// 9 argmin passes; B-fragments cached in LDS after pass 0.
__global__ __launch_bounds__(32)
void atss_topk_wmma(const float* __restrict__ pred,
                    const float* __restrict__ gt,
                    long long* __restrict__ ws_key,
                    int C, int tpc)
{
    __shared__ float2 bcache[CACHE_T * 32];

    const int lane = threadIdx.x;
    const int l16  = lane & 15;
    const int hh   = lane >> 4;

    const int bid   = blockIdx.x;
    const int chunk = bid % C;
    const int gtile = (bid / C) & 3;
    const int b     = bid / (C * 4);
    const int gbase = gtile * 16;

    // A fragment (16x4 f32): lanes 0-15 hold K=0,1 ; lanes 16-31 hold K=2,3
    // row g: [gx, gy, 1, gx^2+gy^2]
    const float* gp = gt + ((size_t)b * G_ + (gbase + l16)) * 4;
    const float gx = gp[0], gy = gp[1];
    v2f a;
    a.x = hh ? 1.0f            : gx;
    a.y = hh ? (gx*gx + gy*gy) : gy;

    const float* pbase = pred + (size_t)b * N_ * 4;

    int t0 = chunk * tpc;
    int t1 = t0 + tpc; if (t1 > TILES_) t1 = TILES_;
    if (t0 > TILES_) t0 = TILES_;
    const int nt  = t1 - t0;
    const int ntc = nt < CACHE_T ? nt : CACHE_T;   // tiles cached in LDS

    long long prev[8], best[8];
#pragma unroll
    for (int v = 0; v < 8; ++v) prev[v] = -1LL;

    for (int p = 0; p < K_; ++p) {
#pragma unroll
        for (int v = 0; v < 8; ++v) best[v] = KEY_MAX;

        if (p == 0) {
            // pass 0: build B fragments from global, cache to LDS
            for (int t = 0; t < nt; ++t) {
                const int n = (t0 + t) * 16 + l16;
                const float2 pc = *(const float2*)(pbase + (size_t)n * 4);
                v2f bb;   // col n: [-2px, -2py, px^2+py^2, 1]
                bb.x = hh ? (pc.x*pc.x + pc.y*pc.y) : (-2.0f * pc.x);
                bb.y = hh ? 1.0f                    : (-2.0f * pc.y);
                if (t < ntc) bcache[t * 32 + lane] = make_float2(bb.x, bb.y);
                tile_update(a, bb, n, prev, best);
            }
        } else {
            // later passes: cached fragments from LDS, remainder from global
            for (int t = 0; t < ntc; ++t) {
                const int n = (t0 + t) * 16 + l16;
                const float2 f = bcache[t * 32 + lane];
                v2f bb; bb.x = f.x; bb.y = f.y;
                tile_update(a, bb, n, prev, best);
            }
            for (int t = ntc; t < nt; ++t) {
                const int n = (t0 + t) * 16 + l16;
                const float2 pc = *(const float2*)(pbase + (size_t)n * 4);
                v2f bb;
                bb.x = hh ? (pc.x*pc.x + pc.y*pc.y) : (-2.0f * pc.x);
                bb.y = hh ? 1.0f                    : (-2.0f * pc.y);
                tile_update(a, bb, n, prev, best);
            }
        }

        // butterfly min within each 16-lane half (xor 8,4,2,1 stays in-half)
#pragma unroll
        for (int v = 0; v < 8; ++v) {
#pragma unroll
            for (int off = 8; off >= 1; off >>= 1) {
                const long long o = __shfl_xor(best[v], off, 32);
                best[v] = (o < best[v]) ? o : best[v];
            }
            prev[v] = best[v];
        }
        if (l16 == 0) {
#pragma unroll
            for (int v = 0; v < 8; ++v) {
                const int g = gbase + v + 8 * hh;
                ws_key[(((size_t)b * G_ + g) * (size_t)C + chunk) * K_ + p] = best[v];
            }
        }
    }
}

// One thread per (b,g): merge C*9 chunk keys -> global top-9, then IoU,
// threshold mean + unbiased std, center-inside mask, write 4 outputs.
__global__ void atss_finalize(const float* __restrict__ pred,
                              const float* __restrict__ gt,
                              const long long* __restrict__ ws_key,
                              float* __restrict__ out, int C)
{
    const int tid = blockIdx.x * blockDim.x + threadIdx.x;
    if (tid >= B_ * G_) return;
    const int b = tid / G_, g = tid % G_;

    const long long* ck = ws_key + ((size_t)b * G_ + g) * (size_t)C * K_;
    const int M = C * K_;

    const float* gp = gt + ((size_t)b * G_ + g) * 4;
    const float gcx = gp[0], gcy = gp[1], gw = gp[2], gh = gp[3];
    const float gx1 = gcx - 0.5f*gw, gy1 = gcy - 0.5f*gh;
    const float gx2 = gcx + 0.5f*gw, gy2 = gcy + 0.5f*gh;
    const float garea = (gx2 - gx1) * (gy2 - gy1);

    int   kidx[K_];
    float kiou[K_];
    int   kin [K_];

    long long pv = -1LL;
    float iousum = 0.f;
    for (int k = 0; k < K_; ++k) {
        long long bv = KEY_MAX;
        for (int m = 0; m < M; ++m) {
            const long long x = ck[m];
            const bool take = (x > pv) & (x < bv);
            bv = take ? x : bv;
        }
        pv = bv;
        const int bi = (int)(unsigned int)(bv & 0xffffffffLL);
        kidx[k] = bi;

        const float* pp = pred + ((size_t)b * N_ + bi) * 4;
        const float cx = pp[0], cy = pp[1], w = pp[2], h = pp[3];
        const float x1 = cx - 0.5f*w, y1 = cy - 0.5f*h;
        const float x2 = cx + 0.5f*w, y2 = cy + 0.5f*h;
        const float ltx = fmaxf(gx1, x1), lty = fmaxf(gy1, y1);
        const float rbx = fminf(gx2, x2), rby = fminf(gy2, y2);
        const float iw = fmaxf(rbx - ltx, 0.f), ih = fmaxf(rby - lty, 0.f);
        const float inter = iw * ih;
        const float iou = inter / (garea + (x2 - x1) * (y2 - y1) - inter);
        kiou[k] = iou; iousum += iou;
        kin[k] = (gx1 <= cx && cx <= gx2 && gy1 <= cy && cy <= gy2) ? 1 : 0;
    }
    const float mean = iousum / (float)K_;
    float var = 0.f;
    for (int k = 0; k < K_; ++k) { const float d = kiou[k] - mean; var += d * d; }
    var *= (1.0f / (K_ - 1));                  // torch.std unbiased (ddof=1)
    const float thr = mean + sqrtf(var);

    float* o_pred = out;
    float* o_gt   = out + (size_t)B_*G_*K_;
    float* o_mask = out + (size_t)2*B_*G_*K_;
    float* o_iou  = out + (size_t)3*B_*G_*K_;
    const size_t base = (size_t)tid * K_;
    for (int k = 0; k < K_; ++k) {
        const bool m = (kiou[k] >= thr) && kin[k];
        o_pred[base + k] = m ? (float)kidx[k] : -1.f;
        o_gt  [base + k] = m ? (float)g       : -1.f;
        o_mask[base + k] = m ? 1.f : 0.f;
        o_iou [base + k] = kiou[k];
    }
}

extern "C" void kernel_launch(void* const* d_in, const int* in_sizes, int n_in,
                              void* d_out, int out_size, void* d_ws, size_t ws_size,
                              hipStream_t stream) {
    const float* pred = (const float*)d_in[0];   // [16,30000,4] cxcywh
    const float* gt   = (const float*)d_in[1];   // [16,64,4]    cxcywh
    float* out = (float*)d_out;

    // per chunk: B*G*K keys of 8 bytes = 73,728 B
    const size_t per_chunk = (size_t)B_ * G_ * K_ * sizeof(long long);
    int C = 15;                                  // ~1.06 MB total; tpc=125<=CACHE_T
    if (ws_size < per_chunk * (size_t)C) {
        C = (int)(ws_size / per_chunk);
        if (C < 1) C = 1;
    }
    const int tpc = (TILES_ + C - 1) / C;

    long long* ws_key = (long long*)d_ws;

    atss_topk_wmma<<<dim3(B_ * 4 * C), dim3(32), 0, stream>>>(
        pred, gt, ws_key, C, tpc);
    atss_finalize<<<dim3((B_ * G_ + 255) / 256), dim3(256), 0, stream>>>(
        pred, gt, ws_key, out, C);
}